// LucidRainsMinimal_360777253455
// MI455X (gfx1250) — compile-verified
//
#include <hip/hip_runtime.h>
#include <math.h>

#define B_    2
#define H_    8
#define G_    2
#define DH_   64
#define BLK_  64
#define BALL_ 256
#define TOPK_ 2
#define DIM_  512
#define NPB_  2048
#define NT_   4096
#define W_    32          // NPB_/BLK_
#define CH_   4096        // BLK_*DH_
#define QKVD_ 768         // (H_+2G_)*DH_
#define NC_   33          // W_+1 (mem token)
#define SCALE_ 0.125f     // DH_^-0.5
#define KC_   32          // GEMM K-chunk staged via TDM

typedef float v2f __attribute__((ext_vector_type(2)));
typedef float v8f __attribute__((ext_vector_type(8)));
typedef unsigned int u32x4 __attribute__((ext_vector_type(4)));
typedef int i32x4 __attribute__((ext_vector_type(4)));
typedef int i32x8 __attribute__((ext_vector_type(8)));

__device__ __forceinline__ v8f wmma4(v2f a, v2f b, v8f c) {
  // D = A(16x4) * B(4x16) + C, fp32, CDNA5 wave32
  return __builtin_amdgcn_wmma_f32_16x16x4_f32(false, a, false, b, (short)0, c, false, false);
}

#if __has_builtin(__builtin_amdgcn_tensor_load_to_lds) && __has_builtin(__builtin_amdgcn_s_wait_tensorcnt)
#define USE_TDM 1
#else
#define USE_TDM 0
#endif

#if USE_TDM
// Issue one 2-D TDM tile load: rows x rowlen fp32 elements, row stride (elements),
// into LDS at byte offset lds_off. pad=1 inserts 1 dword after every 32 dwords
// (gives a 33-float LDS row stride for the A panel -> conflict-free ds reads).
__device__ __forceinline__ void tdm_load_2d(const float* gsrc, unsigned lds_off,
                                            int rows, int rowlen,
                                            unsigned long long row_stride,
                                            unsigned td0, unsigned td1, int pad)
{
  unsigned long long ga = (unsigned long long)(size_t)gsrc;
  u32x4 g0;
  g0[0] = 1u;                                       // count=1, user mode
  g0[1] = lds_off;                                  // lds_addr (bytes)
  g0[2] = (unsigned)(ga & 0xFFFFFFFFu);             // global_addr[31:0]
  g0[3] = (unsigned)((ga >> 32) & 0x01FFFFFFu) | 0x80000000u; // addr[56:32] | type=2
  i32x8 g1;
  unsigned w0 = (2u << 16);                         // data_size = 4 bytes
  if (pad) w0 |= (1u << 20) | (4u << 22);           // pad_enable, interval=32 dw, amount=1 dw
  g1[0] = (int)w0;
  g1[1] = (int)((td0 & 0xFFFFu) << 16);             // tensor_dim0[15:0] @ bits 63:48
  g1[2] = (int)((td0 >> 16) | ((td1 & 0xFFFFu) << 16));   // td0 hi | td1 lo
  g1[3] = (int)((td1 >> 16) | ((unsigned)rowlen << 16));  // td1 hi | tile_dim0
  g1[4] = rows;                                     // tile_dim1 (tile_dim2 = 0)
  g1[5] = (int)(unsigned)(row_stride & 0xFFFFFFFFu);      // tensor_dim0_stride lo
  g1[6] = (int)(unsigned)((row_stride >> 32) & 0xFFFFu);  // stride hi (dim1_stride = 0)
  g1[7] = 0;
  i32x4 z4 = {0, 0, 0, 0};
#if defined(__clang_major__) && (__clang_major__ >= 23)
  i32x8 z8 = {0, 0, 0, 0, 0, 0, 0, 0};
  __builtin_amdgcn_tensor_load_to_lds(g0, g1, z4, z4, z8, 0);
#else
  __builtin_amdgcn_tensor_load_to_lds(g0, g1, z4, z4, 0);
#endif
}
#endif

// ---------------------------------------------------------------- PE + RMSNorm
__global__ void pe_rms_kernel(const float* __restrict__ x, const float* __restrict__ pos,
                              const float* __restrict__ peW, const float* __restrict__ peb,
                              const float* __restrict__ nsc, float* __restrict__ inp)
{
  __shared__ float mean3[3];
  __shared__ float red[4];
  int row = blockIdx.x;                // 0..4095
  int tid = threadIdx.x;               // 128 threads
  if (tid < 3) {
    int blk = row >> 6;
    float s = 0.f;
    for (int r = 0; r < 64; ++r) s += pos[(blk * 64 + r) * 3 + tid];
    mean3[tid] = s * (1.0f / 64.0f);
  }
  __syncthreads();
  float r0 = pos[row*3+0] - mean3[0];
  float r1 = pos[row*3+1] - mean3[1];
  float r2 = pos[row*3+2] - mean3[2];
  float vals[4];
  float ss = 0.f;
#pragma unroll
  for (int u = 0; u < 4; ++u) {
    int c = tid + u * 128;
    float v = x[(size_t)row*DIM_ + c] + r0*peW[c] + r1*peW[DIM_+c] + r2*peW[2*DIM_+c] + peb[c];
    vals[u] = v; ss += v * v;
  }
  for (int off = 16; off > 0; off >>= 1) ss += __shfl_down(ss, off, 32);
  if ((tid & 31) == 0) red[tid >> 5] = ss;
  __syncthreads();
  float tot = red[0] + red[1] + red[2] + red[3];
  float rinv = rsqrtf(tot / (float)DIM_ + 1e-6f);
#pragma unroll
  for (int u = 0; u < 4; ++u) {
    int c = tid + u * 128;
    inp[(size_t)row*DIM_ + c] = vals[u] * rinv * nsc[c];
  }
}

// ------------------------------------------- TDM-fed double-buffered WMMA GEMM
// C[M,N] = A[M,K] @ B[K,N] (+bias)(+relu). 256 threads: 8 waves x (16 rows x 64 cols).
// M % 128 == 0, N % 64 == 0, K % 32 == 0.
__global__ void gemm_kernel(const float* __restrict__ A, const float* __restrict__ Bm,
                            const float* __restrict__ bias, float* __restrict__ C,
                            int M, int N, int K, int relu)
{
  __shared__ float As[2][128][33];     // 33-stride: matches TDM pad (32dw + 1dw)
  __shared__ float Bs[2][KC_][64];
  int lane = threadIdx.x & 31;
  int wave = threadIdx.x >> 5;
  int ntile = N >> 6;
  int mb = blockIdx.x / ntile;
  int nt = blockIdx.x - mb * ntile;
  int rowB = mb << 7;                  // 128-row block
  int col0 = nt << 6;
  int w16  = wave << 4;
  int m  = lane & 15;
  int ks = (lane >> 4) << 1;           // K select: 0 or 2
  v8f a0 = {}, a1 = {}, a2 = {}, a3 = {};
  int nch = K >> 5;

#if USE_TDM
  if (wave == 0) {
    tdm_load_2d(A + (size_t)rowB * K, (unsigned)(size_t)&As[0][0][0],
                128, KC_, (unsigned long long)K, (unsigned)K, (unsigned)M, 1);
    tdm_load_2d(Bm + col0, (unsigned)(size_t)&Bs[0][0][0],
                KC_, 64, (unsigned long long)N, (unsigned)N, (unsigned)K, 0);
  }
  for (int c = 0; c < nch; ++c) {
    int buf = c & 1;
    if (wave == 0) __builtin_amdgcn_s_wait_tensorcnt(0);
    __syncthreads();
    if (wave == 0 && c + 1 < nch) {
      tdm_load_2d(A + (size_t)rowB * K + (size_t)(c + 1) * KC_,
                  (unsigned)(size_t)&As[buf ^ 1][0][0],
                  128, KC_, (unsigned long long)K, (unsigned)K, (unsigned)M, 1);
      tdm_load_2d(Bm + (size_t)(c + 1) * KC_ * N + col0,
                  (unsigned)(size_t)&Bs[buf ^ 1][0][0],
                  KC_, 64, (unsigned long long)N, (unsigned)N, (unsigned)K, 0);
    }
    const float (*Ab)[33] = As[buf];
    const float (*Bb)[64] = Bs[buf];
#pragma unroll
    for (int k0 = 0; k0 < KC_; k0 += 4) {
      v2f a; a.x = Ab[w16 + m][k0 + ks]; a.y = Ab[w16 + m][k0 + ks + 1];
      const float* br0 = &Bb[k0 + ks][0];
      const float* br1 = &Bb[k0 + ks + 1][0];
      v2f b0, b1, b2, b3;
      b0.x = br0[m];      b0.y = br1[m];
      b1.x = br0[m + 16]; b1.y = br1[m + 16];
      b2.x = br0[m + 32]; b2.y = br1[m + 32];
      b3.x = br0[m + 48]; b3.y = br1[m + 48];
      a0 = wmma4(a, b0, a0);
      a1 = wmma4(a, b1, a1);
      a2 = wmma4(a, b2, a2);
      a3 = wmma4(a, b3, a3);
    }
    __syncthreads();
  }
#else
  for (int c = 0; c < nch; ++c) {
    __syncthreads();
    for (int t = threadIdx.x; t < 128 * KC_; t += 256) {
      int r = t >> 5, kk = t & 31;
      As[0][r][kk] = A[(size_t)(rowB + r) * K + (size_t)c * KC_ + kk];
    }
    for (int t = threadIdx.x; t < KC_ * 64; t += 256) {
      int r = t >> 6, nn = t & 63;
      Bs[0][r][nn] = Bm[(size_t)((size_t)c * KC_ + r) * N + col0 + nn];
    }
    __syncthreads();
    const float (*Ab)[33] = As[0];
    const float (*Bb)[64] = Bs[0];
#pragma unroll
    for (int k0 = 0; k0 < KC_; k0 += 4) {
      v2f a; a.x = Ab[w16 + m][k0 + ks]; a.y = Ab[w16 + m][k0 + ks + 1];
      const float* br0 = &Bb[k0 + ks][0];
      const float* br1 = &Bb[k0 + ks + 1][0];
      v2f b0, b1, b2, b3;
      b0.x = br0[m];      b0.y = br1[m];
      b1.x = br0[m + 16]; b1.y = br1[m + 16];
      b2.x = br0[m + 32]; b2.y = br1[m + 32];
      b3.x = br0[m + 48]; b3.y = br1[m + 48];
      a0 = wmma4(a, b0, a0);
      a1 = wmma4(a, b1, a1);
      a2 = wmma4(a, b2, a2);
      a3 = wmma4(a, b3, a3);
    }
  }
#endif

  int rbase = rowB + w16 + ((lane >> 4) << 3);
#pragma unroll
  for (int r = 0; r < 8; ++r) {
    int rr = rbase + r;
    float* Cp = C + (size_t)rr * N + col0 + m;
    float o0 = a0[r], o1 = a1[r], o2 = a2[r], o3 = a3[r];
    if (bias) {
      o0 += bias[col0 + m];      o1 += bias[col0 + 16 + m];
      o2 += bias[col0 + 32 + m]; o3 += bias[col0 + 48 + m];
    }
    if (relu) {
      o0 = fmaxf(o0, 0.f); o1 = fmaxf(o1, 0.f);
      o2 = fmaxf(o2, 0.f); o3 = fmaxf(o3, 0.f);
    }
    Cp[0] = o0; Cp[16] = o1; Cp[32] = o2; Cp[48] = o3;
  }
}

// ------------------------------------------- compression inputs: (k|v)+pos, pre-RoPE
__global__ void build_comp_in(const float* __restrict__ qkv, const float* __restrict__ kpos,
                              const float* __restrict__ vpos, float* __restrict__ kbf,
                              float* __restrict__ vbf)
{
  int t = blockIdx.x * 256 + threadIdx.x;     // B*G*NPB*DH
  if (t >= B_*G_*NPB_*DH_) return;
  int d = t & 63; int r = t >> 6;
  int i = r % NPB_; r /= NPB_;
  int g = r % G_;   int b = r / G_;
  int j = i & 63;
  size_t src = (size_t)(b * NPB_ + i) * QKVD_;
  size_t dst = ((size_t)(b * G_ + g) * NPB_ + i) * DH_ + d;
  kbf[dst] = qkv[src + DIM_ + g*DH_ + d]           + kpos[(g*BLK_ + j)*DH_ + d];
  vbf[dst] = qkv[src + DIM_ + G_*DH_ + g*DH_ + d]  + vpos[(g*BLK_ + j)*DH_ + d];
}

// ---------------------------------------------------------------- RoPE + split
__global__ void rope_split(const float* __restrict__ qkv, float* __restrict__ q_r,
                           float* __restrict__ k_r, float* __restrict__ v_r)
{
  int t = blockIdx.x * 256 + threadIdx.x;     // B*NPB*768
  if (t >= B_*NPB_*QKVD_) return;
  int c = t % QKVD_; int row = t / QKVD_;
  int b = row / NPB_, i = row % NPB_;
  float val = qkv[(size_t)row * QKVD_ + c];
  int d = c & 63;
  if (c < DIM_ + G_*DH_) {
    int j = d >> 1;
    float inv = powf(10000.f, -(float)(2 * j) / (float)DH_);
    float ang = (float)i * inv;
    float cs = cosf(ang), sn = sinf(ang);
    float other = qkv[(size_t)row * QKVD_ + (c ^ 1)];
    float o = (d & 1) ? fmaf(val, cs,  other * sn)
                      : fmaf(val, cs, -other * sn);
    if (c < DIM_) {
      int h = c >> 6;
      q_r[(((size_t)(b*H_ + h)) * NPB_ + i) * DH_ + d] = o;
    } else {
      int g = (c - DIM_) >> 6;
      k_r[(((size_t)(b*G_ + g)) * NPB_ + i) * DH_ + d] = o;
    }
  } else {
    int g = (c - DIM_ - G_*DH_) >> 6;
    v_r[(((size_t)(b*G_ + g)) * NPB_ + i) * DH_ + d] = val;
  }
}

// ------------------------------------------- assemble ck/cv with mem tokens
__global__ void assemble_c(const float* __restrict__ ckb, const float* __restrict__ cvb,
                           const float* __restrict__ memk, const float* __restrict__ memv,
                           float* __restrict__ ck, float* __restrict__ cv)
{
  int t = blockIdx.x * 256 + threadIdx.x;     // B*G*33*64
  if (t >= B_*G_*NC_*DH_) return;
  int d = t & 63; int r = t >> 6;
  int j = r % NC_; r /= NC_;
  int g = r % G_;  int b = r / G_;
  size_t dst = ((size_t)(b*G_ + g) * NC_ + j) * DH_ + d;
  if (j == 0) { ck[dst] = memk[g*DH_ + d]; cv[dst] = memv[g*DH_ + d]; }
  else {
    size_t s = ((size_t)(b*G_ + g) * W_ + (j - 1)) * DH_ + d;
    ck[dst] = ckb[s]; cv[dst] = cvb[s];
  }
}

// ------------------------------------- compressed attention + importance + top-2
__global__ void cattn_kernel(const float* __restrict__ q_r, const float* __restrict__ ck,
                             const float* __restrict__ cv, float* __restrict__ c_out,
                             int* __restrict__ sel_idx, float* __restrict__ sel_val)
{
  __shared__ float sck[NC_ * DH_];
  __shared__ float scv[NC_ * DH_];
  __shared__ float simp[BLK_ * W_];
  int bg = blockIdx.x >> 5;                  // (b,g) pair ; 32 i-tiles each
  int it = blockIdx.x & 31;
  int b = bg >> 1, g = bg & 1;
  for (int t = threadIdx.x; t < NC_*DH_; t += 256) {
    sck[t] = ck[(size_t)bg * NC_ * DH_ + t];
    scv[t] = cv[(size_t)bg * NC_ * DH_ + t];
  }
  for (int t = threadIdx.x; t < BLK_*W_; t += 256) simp[t] = 0.f;
  __syncthreads();
  int h  = threadIdx.x >> 6;                 // 0..3 heads within group
  int il = threadIdx.x & 63;
  int i  = (it << 6) + il;
  int head = g * (H_ / G_) + h;
  const float* qp = q_r + (((size_t)(b*H_ + head)) * NPB_ + i) * DH_;
  float qv[DH_];
#pragma unroll
  for (int d = 0; d < DH_; ++d) qv[d] = qp[d];
  int vis = i >> 6;                          // # fully-preceding blocks (+ mem at j=0)
  float s[NC_];
  float mx = -1e30f;
  for (int j = 0; j <= vis; ++j) {
    float acc = 0.f;
#pragma unroll
    for (int d = 0; d < DH_; ++d) acc = fmaf(qv[d], sck[j*DH_ + d], acc);
    acc *= SCALE_;
    s[j] = acc; mx = fmaxf(mx, acc);
  }
  float l = 0.f;
  for (int j = 0; j <= vis; ++j) { float p = expf(s[j] - mx); s[j] = p; l += p; }
  float linv = 1.f / l;
  float out[DH_];
#pragma unroll
  for (int d = 0; d < DH_; ++d) out[d] = 0.f;
  for (int j = 0; j <= vis; ++j) {
    float p = s[j] * linv;
    if (j >= 1) atomicAdd(&simp[il * W_ + (j - 1)], p * 0.25f);  // mean over 4 heads
#pragma unroll
    for (int d = 0; d < DH_; ++d) out[d] = fmaf(p, scv[j*DH_ + d], out[d]);
  }
  float* cop = c_out + (((size_t)(b*H_ + head)) * NPB_ + i) * DH_;
#pragma unroll
  for (int d = 0; d < DH_; ++d) cop[d] = out[d];
  __syncthreads();
  if (h == 0) {
    float v0 = -1.f, v1 = -1.f; int i0 = 0, i1 = 0;
    for (int w = 0; w < W_; ++w) {
      float v = simp[il * W_ + w];
      if (v > v0)      { v1 = v0; i1 = i0; v0 = v; i0 = w; }
      else if (v > v1) { v1 = v;  i1 = w; }
    }
    size_t o = ((size_t)(b*G_ + g) * NPB_ + i) * TOPK_;
    sel_idx[o] = i0; sel_idx[o+1] = i1;
    sel_val[o] = v0; sel_val[o+1] = v1;
  }
}

// ------------------------------------------------ fine (selected-block) attention
__global__ void fattn_kernel(const float* __restrict__ q_r, const float* __restrict__ k_r,
                             const float* __restrict__ v_r, const int* __restrict__ sel_idx,
                             const float* __restrict__ sel_val, float* __restrict__ f_out)
{
  int t = blockIdx.x * 256 + threadIdx.x;    // B*G*4*NPB threads
  int i = t & (NPB_ - 1); t >>= 11;
  int h = t & 3; t >>= 2;
  int g = t & 1; int b = t >> 1;
  int head = g * (H_ / G_) + h;
  const float* qp = q_r + (((size_t)(b*H_ + head)) * NPB_ + i) * DH_;
  float qv[DH_];
#pragma unroll
  for (int d = 0; d < DH_; ++d) qv[d] = qp[d];
  int own = i >> 6;
  size_t so = ((size_t)(b*G_ + g) * NPB_ + i) * TOPK_;
  int s0 = sel_idx[so], s1 = sel_idx[so + 1];
  bool ok0 = (sel_val[so]     > 0.f) && (s0 != own);
  bool ok1 = (sel_val[so + 1] > 0.f) && (s1 != own);
  int  blks[3] = { s0, s1, own };
  bool bok[3]  = { ok0, ok1, true };
  const float* kbase = k_r + ((size_t)(b*G_ + g)) * NPB_ * DH_;
  const float* vbase = v_r + ((size_t)(b*G_ + g)) * NPB_ * DH_;
  float mx = -1e30f;
  for (int bi = 0; bi < 3; ++bi) {
    if (!bok[bi]) continue;
    int jmax = (bi == 2) ? (i & 63) : 63;
    const float* kp = kbase + (size_t)blks[bi] * BLK_ * DH_;
    for (int jj = 0; jj <= jmax; ++jj) {
      float acc = 0.f;
#pragma unroll
      for (int d = 0; d < DH_; ++d) acc = fmaf(qv[d], kp[jj*DH_ + d], acc);
      mx = fmaxf(mx, acc * SCALE_);
    }
  }
  float l = 0.f;
  float out[DH_];
#pragma unroll
  for (int d = 0; d < DH_; ++d) out[d] = 0.f;
  for (int bi = 0; bi < 3; ++bi) {
    if (!bok[bi]) continue;
    int jmax = (bi == 2) ? (i & 63) : 63;
    const float* kp = kbase + (size_t)blks[bi] * BLK_ * DH_;
    const float* vp = vbase + (size_t)blks[bi] * BLK_ * DH_;
    for (int jj = 0; jj <= jmax; ++jj) {
      float acc = 0.f;
#pragma unroll
      for (int d = 0; d < DH_; ++d) acc = fmaf(qv[d], kp[jj*DH_ + d], acc);
      float p = expf(acc * SCALE_ - mx);
      l += p;
#pragma unroll
      for (int d = 0; d < DH_; ++d) out[d] = fmaf(p, vp[jj*DH_ + d], out[d]);
    }
  }
  float linv = 1.f / l;
  float* fo = f_out + (((size_t)(b*H_ + head)) * NPB_ + i) * DH_;
#pragma unroll
  for (int d = 0; d < DH_; ++d) fo[d] = out[d] * linv;
}

// --------------------------------------------- ball (windowed dense) attn, WMMA
__global__ void ballattn_kernel(const float* __restrict__ q_r, const float* __restrict__ k_r,
                                const float* __restrict__ v_r, float* __restrict__ s_out)
{
  __shared__ float S[2][16][260];            // 2 waves x 16 rows x 256 (+pad)
  int lane = threadIdx.x & 31;
  int wave = threadIdx.x >> 5;
  int gw = blockIdx.x;
  int part = gw & 7;  gw >>= 3;              // 8 parts of 2 q-tiles
  int win  = gw & 7;  gw >>= 3;              // 8 windows of 256
  int h    = gw & 7;  int b = gw >> 3;
  int g = h >> 2;
  int qt = part * 2 + wave;                  // 0..15
  int row0 = win * BALL_ + qt * 16;
  const float* Q = q_r + ((size_t)(b*H_ + h)) * NPB_ * DH_;
  const float* K = k_r + (((size_t)(b*G_ + g)) * NPB_ + win * BALL_) * DH_;
  const float* V = v_r + (((size_t)(b*G_ + g)) * NPB_ + win * BALL_) * DH_;
  int m  = lane & 15;
  int ks = (lane >> 4) << 1;
  int rb = (lane >> 4) << 3;
  // S = scale * Q Kt  (16 x 256)
  for (int ct = 0; ct < 16; ++ct) {
    v8f acc = {};
#pragma unroll
    for (int k0 = 0; k0 < DH_; k0 += 4) {
      v2f a;  a.x  = Q[(row0 + m) * DH_ + k0 + ks];   a.y  = Q[(row0 + m) * DH_ + k0 + ks + 1];
      v2f bb; bb.x = K[(ct*16 + m) * DH_ + k0 + ks];  bb.y = K[(ct*16 + m) * DH_ + k0 + ks + 1];
      acc = wmma4(a, bb, acc);
    }
#pragma unroll
    for (int r = 0; r < 8; ++r) S[wave][rb + r][ct*16 + m] = acc[r] * SCALE_;
  }
  __syncthreads();
  if (lane < 16) {                           // softmax per row
    float* rowp = &S[wave][lane][0];
    float mxv = -1e30f;
    for (int c2 = 0; c2 < BALL_; ++c2) mxv = fmaxf(mxv, rowp[c2]);
    float l = 0.f;
    for (int c2 = 0; c2 < BALL_; ++c2) { float p = expf(rowp[c2] - mxv); rowp[c2] = p; l += p; }
    float linv = 1.f / l;
    for (int c2 = 0; c2 < BALL_; ++c2) rowp[c2] *= linv;
  }
  __syncthreads();
  // O = P V  (16 x 64)
  for (int ct = 0; ct < 4; ++ct) {
    v8f acc = {};
    for (int k0 = 0; k0 < BALL_; k0 += 4) {
      v2f a;  a.x  = S[wave][m][k0 + ks];          a.y  = S[wave][m][k0 + ks + 1];
      v2f bb; bb.x = V[(k0 + ks) * DH_ + ct*16 + m];
      bb.y = V[(k0 + ks + 1) * DH_ + ct*16 + m];
      acc = wmma4(a, bb, acc);
    }
#pragma unroll
    for (int r = 0; r < 8; ++r)
      s_out[(((size_t)(b*H_ + h)) * NPB_ + row0 + rb + r) * DH_ + ct*16 + m] = acc[r];
  }
}

// ---------------------------------------------------------------- gates
__global__ void gates_kernel(const float* __restrict__ inp, const float* __restrict__ Wg,
                             const float* __restrict__ bgv, float* __restrict__ gates)
{
  int t = blockIdx.x * 256 + threadIdx.x;    // NT*24
  if (t >= NT_ * 24) return;
  int c = t % 24; int row = t / 24;
  float acc = bgv[c];
  for (int k = 0; k < DIM_; ++k) acc = fmaf(inp[(size_t)row*DIM_ + k], Wg[(size_t)k*24 + c], acc);
  gates[t] = 1.f / (1.f + expf(-acc));
}

// ---------------------------------------------------------------- gated combine
__global__ void combine_kernel(const float* __restrict__ gates, const float* __restrict__ c_out,
                               const float* __restrict__ f_out, const float* __restrict__ s_out,
                               float* __restrict__ pre)
{
  int t = blockIdx.x * 256 + threadIdx.x;    // NT*DIM
  if (t >= NT_ * DIM_) return;
  int d = t & 63; int h = (t >> 6) & 7; int row = t >> 9;
  int b = row >> 11; int i = row & (NPB_ - 1);
  const float* gp = gates + (size_t)row * 24 + h * 3;
  size_t ho = (((size_t)(b*H_ + h)) * NPB_ + i) * DH_ + d;
  pre[t] = gp[0] * c_out[ho] + gp[1] * f_out[ho] + gp[2] * s_out[ho];
}

// ================================================================ launcher
extern "C" void kernel_launch(void* const* d_in, const int* in_sizes, int n_in,
                              void* d_out, int out_size, void* d_ws, size_t ws_size,
                              hipStream_t stream)
{
  (void)in_sizes; (void)n_in; (void)out_size; (void)ws_size;
  const float* x    = (const float*)d_in[0];
  const float* pos  = (const float*)d_in[1];
  const float* peW  = (const float*)d_in[2];
  const float* peb  = (const float*)d_in[3];
  const float* nsc  = (const float*)d_in[4];
  const float* Wqkv = (const float*)d_in[5];
  const float* kpos = (const float*)d_in[6];
  const float* vpos = (const float*)d_in[7];
  const float* kcW1 = (const float*)d_in[8];
  const float* kcb1 = (const float*)d_in[9];
  const float* kcW2 = (const float*)d_in[10];
  const float* kcb2 = (const float*)d_in[11];
  const float* vcW1 = (const float*)d_in[12];
  const float* vcb1 = (const float*)d_in[13];
  const float* vcW2 = (const float*)d_in[14];
  const float* vcb2 = (const float*)d_in[15];
  const float* memk = (const float*)d_in[16];
  const float* memv = (const float*)d_in[17];
  const float* Wg   = (const float*)d_in[18];
  const float* bgv  = (const float*)d_in[19];
  const float* Wo   = (const float*)d_in[20];
  float* out = (float*)d_out;

  float* p = (float*)d_ws;
  float* inp  = p; p += (size_t)NT_ * DIM_;
  float* qkv  = p; p += (size_t)NT_ * QKVD_;
  float* q_r  = p; p += (size_t)B_*H_*NPB_*DH_;
  float* k_r  = p; p += (size_t)B_*G_*NPB_*DH_;
  float* v_r  = p; p += (size_t)B_*G_*NPB_*DH_;
  float* kbf  = p; p += (size_t)B_*G_*NPB_*DH_;
  float* vbf  = p; p += (size_t)B_*G_*NPB_*DH_;
  float* h1k  = p; p += (size_t)B_*G_*W_*CH_;
  float* h1v  = p; p += (size_t)B_*G_*W_*CH_;
  float* ckb  = p; p += (size_t)B_*G_*W_*DH_;
  float* cvb  = p; p += (size_t)B_*G_*W_*DH_;
  float* ck   = p; p += (size_t)B_*G_*NC_*DH_;
  float* cv   = p; p += (size_t)B_*G_*NC_*DH_;
  float* c_o  = p; p += (size_t)B_*H_*NPB_*DH_;
  float* f_o  = p; p += (size_t)B_*H_*NPB_*DH_;
  float* s_o  = p; p += (size_t)B_*H_*NPB_*DH_;
  float* selv = p; p += (size_t)B_*G_*NPB_*TOPK_;
  float* gat  = p; p += (size_t)NT_ * 24;
  float* pre  = p; p += (size_t)NT_ * DIM_;
  int*   seli = (int*)p;

  auto gemm = [&](const float* A, const float* Bm, const float* bias, float* C,
                  int M, int N, int K, int relu) {
    int blocks = (M >> 7) * (N >> 6);        // 128x64 tile per 256-thread block
    gemm_kernel<<<blocks, 256, 0, stream>>>(A, Bm, bias, C, M, N, K, relu);
  };

  pe_rms_kernel<<<NT_, 128, 0, stream>>>(x, pos, peW, peb, nsc, inp);
  gemm(inp, Wqkv, nullptr, qkv, NT_, QKVD_, DIM_, 0);

  build_comp_in<<<(B_*G_*NPB_*DH_) / 256, 256, 0, stream>>>(qkv, kpos, vpos, kbf, vbf);
  rope_split<<<(B_*NPB_*QKVD_ + 255) / 256, 256, 0, stream>>>(qkv, q_r, k_r, v_r);

  gemm(kbf, kcW1, kcb1, h1k, B_*G_*W_, CH_, CH_, 1);
  gemm(vbf, vcW1, vcb1, h1v, B_*G_*W_, CH_, CH_, 1);
  gemm(h1k, kcW2, kcb2, ckb, B_*G_*W_, DH_, CH_, 0);
  gemm(h1v, vcW2, vcb2, cvb, B_*G_*W_, DH_, CH_, 0);
  assemble_c<<<(B_*G_*NC_*DH_ + 255) / 256, 256, 0, stream>>>(ckb, cvb, memk, memv, ck, cv);

  cattn_kernel<<<B_*G_*(NPB_/BLK_), 256, 0, stream>>>(q_r, ck, cv, c_o, seli, selv);
  fattn_kernel<<<(B_*G_*(H_/G_)*NPB_) / 256, 256, 0, stream>>>(q_r, k_r, v_r, seli, selv, f_o);
  ballattn_kernel<<<B_*H_*(NPB_/BALL_)*8, 64, 0, stream>>>(q_r, k_r, v_r, s_o);

  gates_kernel<<<(NT_*24 + 255) / 256, 256, 0, stream>>>(inp, Wg, bgv, gat);
  combine_kernel<<<(NT_*DIM_) / 256, 256, 0, stream>>>(gat, c_o, f_o, s_o, pre);
  gemm(pre, Wo, nullptr, out, NT_, DIM_, DIM_, 0);
}